// _CRF_15530601742419
// MI455X (gfx1250) — compile-verified
//
#include <hip/hip_runtime.h>
#include <cstdint>

// CRF forward (scaled forward algorithm in linear space) + sequence score + mean.
// B=512, T=1024, K=48. Recurrence per step via V_WMMA_F32_16X16X4_F32:
//   alphaNew(16x48) = alpha(16x48) x expT(48x48), tiled 16x16, K-chunks of 4.
// Latency optimizations: dual WMMA accumulator chains, register-staged
// emission double-buffering, 8-bpermute packed butterfly row reduction,
// 2 barriers per step via parity-buffered rowSum/mask.

#define B_  512
#define T_  1024
#define K_  48
#define BG  16          // batch rows per workgroup (= WMMA M)
#define AST 50          // padded LDS row stride (floats)

typedef __attribute__((ext_vector_type(2))) float v2f;
typedef __attribute__((ext_vector_type(8))) float v8f;

__global__ __launch_bounds__(96) void crf_forward(
    const float* __restrict__ em,        // [B][T][K]
    const uint8_t* __restrict__ mask,    // [B][T] (jax bool -> 1 byte)
    const float* __restrict__ trans,     // [K][K]
    const float* __restrict__ startT,    // [K]
    const float* __restrict__ endT,      // [K]
    float* __restrict__ logZ)            // [B]
{
    __shared__ float alpha[BG * AST];    // normalized forward variables
    __shared__ float alphaN[BG * AST];   // raw next-step values
    __shared__ float emb[BG * AST];      // staged emissions for current step
    __shared__ float rowSumB[2][BG];     // parity-buffered row scales
    __shared__ int   mrowB[2][BG];       // parity-buffered step masks
    __shared__ float cval[BG];           // accumulated log-scale per row

    const int tid  = threadIdx.x;
    const int wave = tid >> 5;           // 0..2
    const int lane = tid & 31;
    const int lrow = lane & 15;
    const int lhi  = lane >> 4;
    const int b0   = blockIdx.x * BG;
    const int j0   = wave * 16;          // this wave's output-column tile

    // ---- loop-invariant B fragments: expT chunks (4x16, K x N) ----
    v2f bfrag[12];
#pragma unroll
    for (int c = 0; c < 12; ++c) {
        const int k = 4 * c + 2 * lhi;
        const int j = j0 + lrow;
        bfrag[c].x = __expf(trans[k * K_ + j]);
        bfrag[c].y = __expf(trans[(k + 1) * K_ + j]);
    }

    // ---- t = 0: alpha0 = exp(start + em0), normalized; c = log(sum) ----
    if (tid < BG) rowSumB[0][tid] = 0.0f;
    __syncthreads();
#pragma unroll
    for (int q = 0; q < 8; ++q) {
        const int idx = tid + 96 * q;
        const int r = idx / K_, j = idx - r * K_;
        const float v = __expf(startT[j] + em[((size_t)(b0 + r) * T_) * K_ + j]);
        alphaN[r * AST + j] = v;
        atomicAdd(&rowSumB[0][r], v);
    }
    __syncthreads();
    {
        const int r = tid & 15, seg = tid >> 4;     // 16 rows x 6 segs of 8 cols
        const float s = rowSumB[0][r];
        const float inv = 1.0f / s;
#pragma unroll
        for (int q = 0; q < 8; ++q) {
            const int j = seg * 8 + q;
            alpha[r * AST + j] = alphaN[r * AST + j] * inv;
        }
        if (seg == 0) cval[r] = __logf(s);
    }
    __syncthreads();

    // ---- stage emissions for t = 1 into registers ----
    float rstage[8];
#pragma unroll
    for (int q = 0; q < 8; ++q) {
        const int idx = tid + 96 * q;
        const int r = idx / K_, j = idx - r * K_;
        rstage[q] = em[((size_t)(b0 + r) * T_ + 1) * K_ + j];
    }

    // ---- recurrence over t (2 barriers / step) ----
    for (int t = 1; t < T_; ++t) {
        const int par = t & 1;

        // Phase A: commit staged emissions to LDS, stage t+1 into registers,
        //          load mask, zero this parity's rowSum.  (Overlaps C of t-1.)
        if (tid < BG)          mrowB[par][tid] = mask[(size_t)(b0 + tid) * T_ + t];
        else if (tid < 2 * BG) rowSumB[par][tid - BG] = 0.0f;
#pragma unroll
        for (int q = 0; q < 8; ++q) {
            const int idx = tid + 96 * q;
            const int r = idx / K_, j = idx - r * K_;
            emb[r * AST + j] = rstage[q];
        }
        if (t + 1 < T_) {
#pragma unroll
            for (int q = 0; q < 8; ++q) {           // in flight through B and C
                const int idx = tid + 96 * q;
                const int r = idx / K_, j = idx - r * K_;
                rstage[q] = em[((size_t)(b0 + r) * T_ + t + 1) * K_ + j];
            }
        }
        if (t + 2 < T_ && tid < BG)                 // warm L2 two steps out
            __builtin_prefetch(&em[((size_t)(b0 + tid) * T_ + t + 2) * K_], 0, 3);
        __syncthreads();

        // Phase B: acc(16x16) = alpha(16x48) x expT(48x16) — two WMMA chains
        v8f acc0 = {}, acc1 = {};
#pragma unroll
        for (int c = 0; c < 12; c += 2) {
            v2f a0, a1;
            const int kb0 = 4 * c + 2 * lhi;
            const int kb1 = kb0 + 4;
            a0.x = alpha[lrow * AST + kb0];
            a0.y = alpha[lrow * AST + kb0 + 1];
            a1.x = alpha[lrow * AST + kb1];
            a1.y = alpha[lrow * AST + kb1 + 1];
            acc0 = __builtin_amdgcn_wmma_f32_16x16x4_f32(
                false, a0, false, bfrag[c],     (short)0, acc0, false, false);
            acc1 = __builtin_amdgcn_wmma_f32_16x16x4_f32(
                false, a1, false, bfrag[c + 1], (short)0, acc1, false, false);
        }
        const v8f acc = acc0 + acc1;

        // emission multiply + stash raw alphaNew
        const int jcol = j0 + lrow;                 // C/D: row = v + 8*lhi, col = lrow
        float val[8];
#pragma unroll
        for (int v = 0; v < 8; ++v) {
            const int r = v + 8 * lhi;
            val[v] = acc[v] * __expf(emb[r * AST + jcol]);
            alphaN[r * AST + jcol] = val[v];
        }

        // packed butterfly: 8 values x 16 lanes -> 8 row sums in 8 shuffles
        float nv[4];
#pragma unroll
        for (int i = 0; i < 4; ++i) {
            const float wire = (lane & 1) ? val[i] : val[i + 4];
            const float recv = __shfl_xor(wire, 1, 16);
            nv[i] = ((lane & 1) ? val[i + 4] : val[i]) + recv;
        }
        float m2[2];
#pragma unroll
        for (int i = 0; i < 2; ++i) {
            const float wire = (lane & 2) ? nv[i] : nv[i + 2];
            const float recv = __shfl_xor(wire, 2, 16);
            m2[i] = ((lane & 2) ? nv[i + 2] : nv[i]) + recv;
        }
        {
            const float wire = (lane & 4) ? m2[0] : m2[1];
            const float recv = __shfl_xor(wire, 4, 16);
            float f = ((lane & 4) ? m2[1] : m2[0]) + recv;
            f += __shfl_xor(f, 8, 16);
            if ((lane & 15) < 8) {                  // one lane per (row, half)
                const int r8 = 4 * (lane & 1) + 2 * ((lane >> 1) & 1)
                             + ((lane >> 2) & 1) + 8 * lhi;
                atomicAdd(&rowSumB[par][r8], f);    // 3 waves -> full 48-col sum
            }
        }
        __syncthreads();

        // Phase C: masked commit — normalize + accumulate log-scale.
        // Runs concurrently with next iteration's Phase A (parity buffers).
        {
            const int r = tid & 15, seg = tid >> 4;
            if (mrowB[par][r]) {
                const float s = rowSumB[par][r];
                const float inv = 1.0f / s;
#pragma unroll
                for (int q = 0; q < 8; ++q) {
                    const int j = seg * 8 + q;
                    alpha[r * AST + j] = alphaN[r * AST + j] * inv;
                }
                if (seg == 0) cval[r] += __logf(s);
            }
        }
    }
    __syncthreads();

    // ---- finalize: logZ = c + log(sum_j alpha[j] * exp(end[j])) ----
    if (tid < BG) {
        float accv = 0.0f;
        for (int j = 0; j < K_; ++j)
            accv += alpha[tid * AST + j] * __expf(endT[j]);
        logZ[b0 + tid] = cval[tid] + __logf(accv);
    }
}

// ---- sequence score: one wave32 per sequence ----
__global__ __launch_bounds__(32) void crf_seqscore(
    const float* __restrict__ em, const int* __restrict__ tags,
    const uint8_t* __restrict__ mask, const float* __restrict__ trans,
    const float* __restrict__ startT, const float* __restrict__ endT,
    float* __restrict__ seqScore)
{
    const int b = blockIdx.x;
    const int lane = threadIdx.x;
    float part = 0.0f;
    int cnt = 0;
    for (int t = 1 + lane; t < T_; t += 32) {
        const int tp = tags[(size_t)b * T_ + t - 1];
        const int tc = tags[(size_t)b * T_ + t];
        const float m = mask[(size_t)b * T_ + t] ? 1.0f : 0.0f;
        part += m * (trans[tp * K_ + tc] + em[((size_t)b * T_ + t) * K_ + tc]);
    }
    for (int t = lane; t < T_; t += 32)
        cnt += mask[(size_t)b * T_ + t] ? 1 : 0;
#pragma unroll
    for (int w = 16; w; w >>= 1) {
        part += __shfl_xor(part, w, 32);
        cnt  += __shfl_xor(cnt,  w, 32);
    }
    if (lane == 0) {
        const int t0 = tags[(size_t)b * T_];
        const int lt = tags[(size_t)b * T_ + (cnt - 1)];
        seqScore[b] = startT[t0] + em[((size_t)b * T_) * K_ + t0] + part + endT[lt];
    }
}

// ---- mean(logZ - seqScore) ----
__global__ __launch_bounds__(256) void crf_reduce(
    const float* __restrict__ logZ, const float* __restrict__ seqScore,
    float* __restrict__ out)
{
    __shared__ float red[8];
    const int tid = threadIdx.x;
    float s = 0.0f;
    for (int i = tid; i < B_; i += 256) s += logZ[i] - seqScore[i];
#pragma unroll
    for (int w = 16; w; w >>= 1) s += __shfl_xor(s, w, 32);
    if ((tid & 31) == 0) red[tid >> 5] = s;
    __syncthreads();
    if (tid < 8) {
        float v = red[tid];
#pragma unroll
        for (int w = 4; w; w >>= 1) v += __shfl_xor(v, w, 8);
        if (tid == 0) out[0] = v / (float)B_;
    }
}

extern "C" void kernel_launch(void* const* d_in, const int* in_sizes, int n_in,
                              void* d_out, int out_size, void* d_ws, size_t ws_size,
                              hipStream_t stream) {
    const float*   em     = (const float*)d_in[0];
    const int*     tags   = (const int*)d_in[1];
    const uint8_t* mask   = (const uint8_t*)d_in[2];   // jax bool -> 1 byte
    const float*   trans  = (const float*)d_in[3];
    const float*   startT = (const float*)d_in[4];
    const float*   endT   = (const float*)d_in[5];

    float* logZ     = (float*)d_ws;          // [512]
    float* seqScore = logZ + B_;             // [512]

    crf_forward <<<B_ / BG, 96,  0, stream>>>(em, mask, trans, startT, endT, logZ);
    crf_seqscore<<<B_,      32,  0, stream>>>(em, tags, mask, trans, startT, endT, seqScore);
    crf_reduce  <<<1,       256, 0, stream>>>(logZ, seqScore, (float*)d_out);
}